// RegionalDecoder_90305982366364
// MI455X (gfx1250) — compile-verified
//
#include <hip/hip_runtime.h>
#include <stdint.h>

#define N_MESH_  40962
#define N_GRID_  100000
#define N_EDGES_ 400000
#define D_IN_    256
#define D_HID_   256
#define D_OUT_   128

typedef float v2f __attribute__((ext_vector_type(2)));
typedef float v8f __attribute__((ext_vector_type(8)));
typedef int   v4i __attribute__((ext_vector_type(4)));

#define AS1 __attribute__((address_space(1)))
#define AS3 __attribute__((address_space(3)))

#if __has_builtin(__builtin_amdgcn_global_load_async_to_lds_b128)
#define RD_ASYNC_STAGE 1
#endif

// generic -> addrspace pointer casts via integer round-trip.
// LDS offsets live in the low 32 bits of a generic shared-aperture address;
// global flat addresses are identity-mapped to AS1.
__device__ __forceinline__ AS3 v4i* rd_lds(void* p) {
    return (AS3 v4i*)(uint32_t)(uintptr_t)p;
}
__device__ __forceinline__ AS1 v4i* rd_gbl(const void* p) {
    return (AS1 v4i*)(uintptr_t)p;
}

// ---------------------------------------------------------------------------
// Kernel 1: zero the scatter accumulators (sums[100000][256] + cnt[100000])
// ---------------------------------------------------------------------------
__global__ void rd_zero_ws(float4* __restrict__ sums4, float* __restrict__ cnt) {
    const size_t stride = (size_t)gridDim.x * blockDim.x;
    const size_t total4 = (size_t)N_GRID_ * (D_IN_ / 4);
    for (size_t i = (size_t)blockIdx.x * blockDim.x + threadIdx.x; i < total4; i += stride)
        sums4[i] = make_float4(0.f, 0.f, 0.f, 0.f);
    for (size_t i = (size_t)blockIdx.x * blockDim.x + threadIdx.x; i < N_GRID_; i += stride)
        cnt[i] = 0.f;
}

// ---------------------------------------------------------------------------
// Kernel 2: edge scatter. One wave per edge-iteration: 32 lanes move a 256-f32
// mesh row (2x float4 per lane) and atomically accumulate into the grid sums.
// mesh (42 MB) and sums (102 MB) are both L2-resident (192 MB), so this is
// L2 atomic bandwidth, not HBM.
// ---------------------------------------------------------------------------
__global__ void rd_scatter(const float* __restrict__ mesh,
                           const int*   __restrict__ eidx,
                           float* __restrict__ sums,
                           float* __restrict__ cnt) {
    const int lane = threadIdx.x & 31;
    const int gw   = blockIdx.x * (blockDim.x >> 5) + (threadIdx.x >> 5);
    const int nw   = gridDim.x * (blockDim.x >> 5);
    for (int e = gw; e < N_EDGES_; e += nw) {
        const int src = eidx[e];
        const int dst = eidx[N_EDGES_ + e];
        const float4* mrow = (const float4*)(mesh + (size_t)src * D_IN_);
        float4 v0 = mrow[lane];
        float4 v1 = mrow[lane + 32];
        float* srow = sums + (size_t)dst * D_IN_ + lane * 4;
        atomicAdd(srow + 0, v0.x);
        atomicAdd(srow + 1, v0.y);
        atomicAdd(srow + 2, v0.z);
        atomicAdd(srow + 3, v0.w);
        atomicAdd(srow + 128 + 0, v1.x);
        atomicAdd(srow + 128 + 1, v1.y);
        atomicAdd(srow + 128 + 2, v1.z);
        atomicAdd(srow + 128 + 3, v1.w);
        if (lane == 0) atomicAdd(cnt + dst, 1.0f);
    }
}

// ---------------------------------------------------------------------------
// Kernel 3: fused mean + MLP using V_WMMA_F32_16X16X4_F32.
// 4 waves / block, each wave owns a 16-row tile staged in a 16 KB LDS slice
// (async global->LDS copy, ASYNCcnt-tracked). Mean division is folded into
// the A-fragment load of GEMM1.
//   A frag (16x4 f32): lane(l): M=l%16, holds A[M][k0 + 2*(l/16) + {0,1}]
//   B frag (4x16 f32): lane(l): N=l%16, holds B[k0 + 2*(l/16) + {0,1}][N]
//   C/D   (16x16 f32): vgpr v, lane l: M = v + 8*(l/16), N = l%16
// ---------------------------------------------------------------------------
__global__ __launch_bounds__(128, 1)
void rd_mlp_wmma(const float* __restrict__ W1, const float* __restrict__ b1,
                 const float* __restrict__ W2, const float* __restrict__ b2,
                 const float* __restrict__ sums, const float* __restrict__ cnt,
                 float* __restrict__ out) {
    __shared__ float hA[4][16 * D_IN_];   // 4 waves * 16KB = 64KB static LDS

    const int lane    = threadIdx.x & 31;
    const int wave    = threadIdx.x >> 5;
    const int hi      = lane >> 4;        // 0 | 1
    const int lo      = lane & 15;        // M or N index within tile
    const int rowbase = blockIdx.x * 64 + wave * 16;

    float* A = &hA[wave][0];

    // warm weight cachelines (global_prefetch_b8)
    __builtin_prefetch(W1 + lane * 64, 0, 0);
    __builtin_prefetch(W2 + lane * 64, 0, 0);

    // ---- stage raw sums[16 rows][256] into the wave's LDS slice -----------
#ifdef RD_ASYNC_STAGE
    for (int r = 0; r < 16; ++r) {
        int row = rowbase + r;
        if (row >= N_GRID_) row = N_GRID_ - 1;            // clamp; tail masked at store
        const float* g0 = sums + (size_t)row * D_IN_ + lane * 4;
        float*       l0 = A + r * D_IN_ + lane * 4;
        __builtin_amdgcn_global_load_async_to_lds_b128(rd_gbl(g0),       rd_lds(l0),       0, 0);
        __builtin_amdgcn_global_load_async_to_lds_b128(rd_gbl(g0 + 128), rd_lds(l0 + 128), 0, 0);
    }
#if __has_builtin(__builtin_amdgcn_s_wait_asynccnt)
    __builtin_amdgcn_s_wait_asynccnt(0);
#else
    asm volatile("s_wait_asynccnt 0x0" ::: "memory");
#endif
#else
    for (int r = 0; r < 16; ++r) {
        int row = rowbase + r;
        if (row >= N_GRID_) row = N_GRID_ - 1;
        const float4* srow = (const float4*)(sums + (size_t)row * D_IN_);
        float4 x0 = srow[lane];
        float4 x1 = srow[lane + 32];
        float4* arow = (float4*)(A + r * D_IN_);
        arow[lane]      = x0;
        arow[lane + 32] = x1;
    }
#endif

    // per-lane mean reciprocal: lane's A-fragment row is M = lo (fixed)
    int mrow = rowbase + lo;
    if (mrow >= N_GRID_) mrow = N_GRID_ - 1;
    const float rcl = 1.0f / fmaxf(cnt[mrow], 1.0f);

    // ---- GEMM1: h = silu((sums/cnt) @ W1 + b1), 16 N-tiles in registers ---
    v8f acc[16];
#pragma unroll
    for (int nt = 0; nt < 16; ++nt) {
        const float bv = b1[nt * 16 + lo];                // bias depends only on N
#pragma unroll
        for (int v = 0; v < 8; ++v) acc[nt][v] = bv;
    }

    for (int k0 = 0; k0 < D_IN_; k0 += 4) {
        v2f a;
        const float* ap = A + lo * D_IN_ + k0 + 2 * hi;
        a.x = ap[0] * rcl;                                // fold mean division
        a.y = ap[1] * rcl;
        const int kb = (k0 + 2 * hi) * D_HID_;
#pragma unroll
        for (int nt = 0; nt < 16; ++nt) {
            const int col = nt * 16 + lo;
            v2f b;
            b.x = W1[kb + col];
            b.y = W1[kb + D_HID_ + col];
            acc[nt] = __builtin_amdgcn_wmma_f32_16x16x4_f32(
                false, a, false, b, (short)0, acc[nt], false, false);
        }
    }

    // ---- SiLU, then overwrite LDS slice with h (all A reads complete) -----
#pragma unroll
    for (int nt = 0; nt < 16; ++nt) {
#pragma unroll
        for (int v = 0; v < 8; ++v) {
            const float x = acc[nt][v];
            const float s = x / (1.0f + __expf(-x));
            A[(v + 8 * hi) * D_IN_ + nt * 16 + lo] = s;
        }
    }

    // ---- GEMM2: out = h @ W2 + b2 -----------------------------------------
    v8f acc2[8];
#pragma unroll
    for (int nt = 0; nt < 8; ++nt) {
        const float bv = b2[nt * 16 + lo];
#pragma unroll
        for (int v = 0; v < 8; ++v) acc2[nt][v] = bv;
    }

    for (int k0 = 0; k0 < D_HID_; k0 += 4) {
        v2f a;
        const float* ap = A + lo * D_IN_ + k0 + 2 * hi;   // h is already final
        a.x = ap[0];
        a.y = ap[1];
        const int kb = (k0 + 2 * hi) * D_OUT_;
#pragma unroll
        for (int nt = 0; nt < 8; ++nt) {
            const int col = nt * 16 + lo;
            v2f b;
            b.x = W2[kb + col];
            b.y = W2[kb + D_OUT_ + col];
            acc2[nt] = __builtin_amdgcn_wmma_f32_16x16x4_f32(
                false, a, false, b, (short)0, acc2[nt], false, false);
        }
    }

    // ---- transpose 16x128 tile through LDS -> coalesced b128 stores -------
#pragma unroll
    for (int nt = 0; nt < 8; ++nt) {
#pragma unroll
        for (int v = 0; v < 8; ++v)
            A[(v + 8 * hi) * D_OUT_ + nt * 16 + lo] = acc2[nt][v];
    }
    for (int r = 0; r < 16; ++r) {
        const int row = rowbase + r;
        if (row < N_GRID_) {
            const float4 val = ((const float4*)(A + r * D_OUT_))[lane];
            ((float4*)(out + (size_t)row * D_OUT_))[lane] = val;
        }
    }
}

// ---------------------------------------------------------------------------
extern "C" void kernel_launch(void* const* d_in, const int* in_sizes, int n_in,
                              void* d_out, int out_size, void* d_ws, size_t ws_size,
                              hipStream_t stream) {
    (void)in_sizes; (void)n_in; (void)out_size; (void)ws_size;

    const float* mesh = (const float*)d_in[0];
    const float* W1   = (const float*)d_in[1];
    const float* b1   = (const float*)d_in[2];
    const float* W2   = (const float*)d_in[3];
    const float* b2   = (const float*)d_in[4];
    const int*   eidx = (const int*)d_in[5];
    // d_in[6] = n_grid_nodes scalar (compile-time constant here)

    float* sums = (float*)d_ws;                                // 100000*256 f32
    float* cnt  = sums + (size_t)N_GRID_ * D_IN_;              // 100000 f32

    float* out = (float*)d_out;

    rd_zero_ws<<<8192, 256, 0, stream>>>((float4*)sums, cnt);
    rd_scatter<<<4096, 256, 0, stream>>>(mesh, eidx, sums, cnt);
    rd_mlp_wmma<<<(N_GRID_ + 63) / 64, 128, 0, stream>>>(W1, b1, W2, b2, sums, cnt, out);
}